// GINLayer_12506944766436
// MI455X (gfx1250) — compile-verified
//
#include <hip/hip_runtime.h>
#include <hip/hip_bf16.h>

#define N_NODES   100000
#define N_EDGES   1600000
#define NODE_DIM  64
#define HIDDEN    256
#define OUT_DIM   64
#define BN_EPS    1e-5f

typedef __attribute__((ext_vector_type(2))) float v2f;
typedef __attribute__((ext_vector_type(8))) float v8f;

#define WMMA_F32X4(ACC, A, B) \
  ACC = __builtin_amdgcn_wmma_f32_16x16x4_f32(false, (A), false, (B), (short)0, (ACC), false, false)

// ---------------------------------------------------------------- zero ws
__global__ __launch_bounds__(256) void zero_ws(float* __restrict__ p, int n) {
  int i = blockIdx.x * 256 + threadIdx.x;
  if (i < n) p[i] = 0.0f;
}

// ------------------------------------------------------- edge scatter-add
// tid -> (edge, channel). A wave32 covers 32 consecutive channels of one
// edge: coalesced 128B load of h[src] and 32 consecutive-address atomics.
__global__ __launch_bounds__(256) void edge_scatter(
    const float* __restrict__ h, const int* __restrict__ src,
    const int* __restrict__ dst, float* __restrict__ agg,
    float* __restrict__ deg) {
  unsigned tid = blockIdx.x * 256u + threadIdx.x;
  if (tid >= (unsigned)N_EDGES * NODE_DIM) return;
  int e = tid >> 6;        // /64
  int c = tid & 63;
  int s = src[e];
  int d = dst[e];
  atomicAdd(&agg[d * NODE_DIM + c], h[s * NODE_DIM + c]);
  if (c == 0) atomicAdd(&deg[d], 1.0f);
}

// ------------------------------------------------ fused MLP (WMMA f32)
// Block = 128 threads (4 waves), computes 16 rows of nodes.
// GEMM1: X(16x64) @ W1(64x256) -> hidden in LDS (wave w: cols [64w,64w+64))
// GEMM2: H(16x256) @ W2(256x64) -> out      (wave w: cols [16w,16w+16))
__global__ __launch_bounds__(128) void fused_mlp(
    const float* __restrict__ h,  const float* __restrict__ W1,
    const float* __restrict__ b1, const float* __restrict__ W2,
    const float* __restrict__ b2, const float* __restrict__ agg,
    const float* __restrict__ deg, float* __restrict__ out) {
  __shared__ float xs[16][NODE_DIM + 1];   // +1 pad: conflict-free column reads
  __shared__ float hs[16][HIDDEN + 1];

  const int m0   = blockIdx.x * 16;
  const int tid  = threadIdx.x;
  const int wave = tid >> 5;
  const int lane = tid & 31;
  const int half = lane >> 4;      // 0: K=k..k+1 / M=0..7 ; 1: K=k+2..k+3 / M=8..15
  const int r    = lane & 15;

  // stage x = agg/max(deg,1) + h for rows m0..m0+15
  for (int i = tid; i < 16 * NODE_DIM; i += 128) {
    int m = i >> 6, c = i & 63;
    int g = (m0 + m) * NODE_DIM + c;
    float d = deg[m0 + m];
    d = d > 1.0f ? d : 1.0f;
    xs[m][c] = agg[g] / d + h[g];
  }
  __syncthreads();

  // ---- GEMM1: wave handles 4 column tiles of 16
  v8f acc0 = {}, acc1 = {}, acc2 = {}, acc3 = {};
  const int nb = wave * 64;
  for (int k = 0; k < NODE_DIM; k += 4) {
    v2f a;
    a.x = xs[r][k + 2 * half];
    a.y = xs[r][k + 2 * half + 1];
    const float* w1p0 = W1 + (k + 2 * half) * HIDDEN + nb + r;
    const float* w1p1 = W1 + (k + 2 * half + 1) * HIDDEN + nb + r;
    v2f b;
    b.x = w1p0[0];  b.y = w1p1[0];  WMMA_F32X4(acc0, a, b);
    b.x = w1p0[16]; b.y = w1p1[16]; WMMA_F32X4(acc1, a, b);
    b.x = w1p0[32]; b.y = w1p1[32]; WMMA_F32X4(acc2, a, b);
    b.x = w1p0[48]; b.y = w1p1[48]; WMMA_F32X4(acc3, a, b);
  }
  // bias + relu -> hidden LDS. VGPR j, lane: row j+8*half, col nb+t*16+r
#define STORE_TILE(ACC, T)                                        \
  {                                                               \
    const int n = nb + (T) * 16 + r;                              \
    const float bias = b1[n];                                     \
    _Pragma("unroll") for (int j = 0; j < 8; ++j) {               \
      float v = (ACC)[j] + bias;                                  \
      hs[j + 8 * half][n] = v > 0.0f ? v : 0.0f;                  \
    }                                                             \
  }
  STORE_TILE(acc0, 0)
  STORE_TILE(acc1, 1)
  STORE_TILE(acc2, 2)
  STORE_TILE(acc3, 3)
#undef STORE_TILE
  __syncthreads();

  // ---- GEMM2: one 16x16 output tile per wave
  v8f occ = {};
  const int ob = wave * 16;
  for (int k = 0; k < HIDDEN; k += 4) {
    v2f a;
    a.x = hs[r][k + 2 * half];
    a.y = hs[r][k + 2 * half + 1];
    v2f b;
    b.x = W2[(k + 2 * half) * OUT_DIM + ob + r];
    b.y = W2[(k + 2 * half + 1) * OUT_DIM + ob + r];
    WMMA_F32X4(occ, a, b);
  }
  {
    const float bias = b2[ob + r];
#pragma unroll
    for (int j = 0; j < 8; ++j) {
      float v = occ[j] + bias;
      v = v > 0.0f ? v : 0.0f;   // final relu (pre-BN)
      out[(m0 + j + 8 * half) * OUT_DIM + ob + r] = v;
    }
  }
}

// ----------------------------------------------------- BN statistics
__global__ __launch_bounds__(256) void bn_stats(
    const float* __restrict__ x, float* __restrict__ gsum,
    float* __restrict__ gsq) {
  __shared__ float ssum[OUT_DIM], ssq[OUT_DIM];
  const int tid = threadIdx.x;
  if (tid < OUT_DIM) { ssum[tid] = 0.0f; ssq[tid] = 0.0f; }
  __syncthreads();
  // stride is a multiple of 64 -> each thread owns one fixed channel
  float s = 0.0f, q = 0.0f;
  const int total = N_NODES * OUT_DIM;
  const int stride = gridDim.x * 256;
  for (int i = blockIdx.x * 256 + tid; i < total; i += stride) {
    float v = x[i];
    s += v;
    q += v * v;
  }
  const int c = tid & 63;
  atomicAdd(&ssum[c], s);
  atomicAdd(&ssq[c], q);
  __syncthreads();
  if (tid < OUT_DIM) {
    atomicAdd(&gsum[tid], ssum[tid]);
    atomicAdd(&gsq[tid], ssq[tid]);
  }
}

// ----------------------------------------------------- BN apply (in place)
__global__ __launch_bounds__(256) void bn_apply(
    float* __restrict__ x, const float* __restrict__ gsum,
    const float* __restrict__ gsq, const float* __restrict__ gamma,
    const float* __restrict__ beta) {
  int i = blockIdx.x * 256 + threadIdx.x;
  if (i >= N_NODES * OUT_DIM) return;
  int c = i & 63;
  const float inv_n = 1.0f / (float)N_NODES;
  float mean = gsum[c] * inv_n;
  float var  = gsq[c] * inv_n - mean * mean;
  float sc   = gamma[c] * __frsqrt_rn(var + BN_EPS);
  x[i] = (x[i] - mean) * sc + beta[c];
}

// ---------------------------------------------------------------- launch
extern "C" void kernel_launch(void* const* d_in, const int* in_sizes, int n_in,
                              void* d_out, int out_size, void* d_ws, size_t ws_size,
                              hipStream_t stream) {
  const float* h     = (const float*)d_in[0];
  const float* W1    = (const float*)d_in[1];
  const float* b1    = (const float*)d_in[2];
  const float* W2    = (const float*)d_in[3];
  const float* b2    = (const float*)d_in[4];
  const float* gamma = (const float*)d_in[5];
  const float* beta  = (const float*)d_in[6];
  const int*   src   = (const int*)d_in[7];
  const int*   dst   = (const int*)d_in[8];
  float* out = (float*)d_out;

  // workspace layout (floats): agg[N*64] | deg[N] | gsum[64] | gsq[64]
  float* agg  = (float*)d_ws;
  float* deg  = agg + (size_t)N_NODES * NODE_DIM;
  float* gsum = deg + N_NODES;
  float* gsq  = gsum + OUT_DIM;

  const int nzero = N_NODES * NODE_DIM + N_NODES + 2 * OUT_DIM;
  zero_ws<<<(nzero + 255) / 256, 256, 0, stream>>>((float*)d_ws, nzero);

  const unsigned nscatter = (unsigned)N_EDGES * NODE_DIM;
  edge_scatter<<<(nscatter + 255) / 256, 256, 0, stream>>>(h, src, dst, agg, deg);

  fused_mlp<<<N_NODES / 16, 128, 0, stream>>>(h, W1, b1, W2, b2, agg, deg, out);

  bn_stats<<<1024, 256, 0, stream>>>(out, gsum, gsq);

  bn_apply<<<(N_NODES * OUT_DIM + 255) / 256, 256, 0, stream>>>(out, gsum, gsq, gamma, beta);
}